// Cross_88751204204936
// MI455X (gfx1250) — compile-verified
//
#include <hip/hip_runtime.h>

typedef __attribute__((ext_vector_type(2))) float v2f;
typedef __attribute__((ext_vector_type(8))) float v8f;

#define DIMS 256
#define NLAYERS 4
#define ROWS_PER_WAVE 16
#define WAVES_PER_BLOCK 8
#define ROWS_PER_BLOCK (ROWS_PER_WAVE * WAVES_PER_BLOCK)  // 128

// DCN cross stack, algebraically collapsed:
//   x_l = a_l * x_0 + u_l
//   u_{l+1} = u_l + bias_l + b_lin_l           (row independent, per block)
//   g_l     = u_l . w_l                         (row independent scalars)
//   d_l     = x_0 . w_l                         (per row, WMMA f32 16x16x4)
//   a_{l+1} = a_l * (1 + d_l) + g_l,  a_0 = 1
//   out     = a_L * x_0 + u_L
__global__ void __launch_bounds__(256)
dcn_cross_wmma(const float* __restrict__ x,
               const float* __restrict__ W,
               const float* __restrict__ b_lin,
               const float* __restrict__ bias,
               float* __restrict__ out)
{
    // Zero-padded W: rows 0..3 = W, rows 4..15 = 0 -> branchless B fragments.
    __shared__ float wpad[16 * DIMS];                       // 16 KB
    __shared__ float redw[WAVES_PER_BLOCK];
    __shared__ float u_s[DIMS];
    __shared__ float d_s[WAVES_PER_BLOCK][ROWS_PER_WAVE][NLAYERS];

    const int t    = threadIdx.x;
    const int lane = t & 31;
    const int wave = t >> 5;

    // ---- Stage padded W into LDS (all 8 waves share one B operand) ----
    #pragma unroll
    for (int i = 0; i < 16; ++i)
        wpad[i * DIMS + t] = (i < NLAYERS) ? W[i * DIMS + t] : 0.0f;

    // ---- Per-block precompute of u_L (D-vector) and g_0..3 (scalars) ----
    float g[NLAYERS];
    {
        float u = 0.0f;
        #pragma unroll
        for (int i = 0; i < NLAYERS; ++i) {
            // g_i = dot(u_i, W[i]) : wave32 butterfly, then cross-wave combine
            float part = u * W[i * DIMS + t];
            #pragma unroll
            for (int off = 16; off > 0; off >>= 1)
                part += __shfl_xor(part, off, 32);
            if (lane == 0) redw[wave] = part;
            __syncthreads();
            float gi = 0.0f;
            #pragma unroll
            for (int wv = 0; wv < WAVES_PER_BLOCK; ++wv) gi += redw[wv];
            g[i] = gi;
            __syncthreads();
            u += bias[i * DIMS + t] + b_lin[i];
        }
        u_s[t] = u;  // u_L for column t
    }
    __syncthreads();  // wpad + u_s visible to all waves

    // ---- Phase 1: d = x_tile @ W^T via V_WMMA_F32_16X16X4_F32 ----
    // A (16x4 f32): lane holds M = lane&15; VGPR0/1 hold K = koff, koff+1
    //               with koff = (lane>=16) ? 2 : 0  -> contiguous float2.
    // B (4x16 f32): lane holds N = lane&15; same K split; rows >=4 are the
    //               LDS zero padding, so the load is unconditional.
    const int rowBase = blockIdx.x * ROWS_PER_BLOCK + wave * ROWS_PER_WAVE;
    const int m       = lane & 15;
    const int koff    = (lane >> 4) << 1;

    const float* __restrict__ xrow  = x + (size_t)(rowBase + m) * DIMS + koff;
    const float* __restrict__ brow  = wpad + m * DIMS + koff;

    v8f c = {};
    #pragma unroll 8
    for (int k0 = 0; k0 < DIMS; k0 += 4) {
        v2f a;
        a.x = xrow[k0];
        a.y = xrow[k0 + 1];
        v2f b;
        b.x = brow[k0];
        b.y = brow[k0 + 1];
        // 8 args: (neg_a, A, neg_b, B, c_mod, C, reuse_a, reuse_b)
        c = __builtin_amdgcn_wmma_f32_16x16x4_f32(
                false, a, false, b, (short)0, c, false, false);
    }

    // C/D layout: lane N = lane&15; VGPR j holds M = j + ((lane>=16)?8:0).
    // Columns 0..3 of C are d_0..d_3 for this wave's 16 rows -> stage via LDS.
    if (m < NLAYERS) {
        const int mb = (lane >> 4) * 8;
        #pragma unroll
        for (int j = 0; j < 8; ++j)
            d_s[wave][mb + j][m] = c[j];
    }
    __syncthreads();

    // ---- Phase 2: per-row recurrence + fused scale/add writeout ----
    #pragma unroll 2
    for (int mm = 0; mm < ROWS_PER_WAVE; ++mm) {
        float a = 1.0f;
        #pragma unroll
        for (int i = 0; i < NLAYERS; ++i)
            a = a * (1.0f + d_s[wave][mm][i]) + g[i];

        const size_t row = (size_t)(rowBase + mm);
        const float4* __restrict__ xr = (const float4*)(x + row * DIMS);
        float4* __restrict__ orow     = (float4*)(out + row * DIMS);
        const float4* us              = (const float4*)u_s;

        // 256 cols = 2 chunks of (32 lanes x float4); x re-read hits L2.
        float4 xv0 = xr[lane];
        float4 xv1 = xr[lane + 32];
        float4 uv0 = us[lane];
        float4 uv1 = us[lane + 32];
        float4 o0, o1;
        o0.x = fmaf(a, xv0.x, uv0.x); o0.y = fmaf(a, xv0.y, uv0.y);
        o0.z = fmaf(a, xv0.z, uv0.z); o0.w = fmaf(a, xv0.w, uv0.w);
        o1.x = fmaf(a, xv1.x, uv1.x); o1.y = fmaf(a, xv1.y, uv1.y);
        o1.z = fmaf(a, xv1.z, uv1.z); o1.w = fmaf(a, xv1.w, uv1.w);
        orow[lane]      = o0;
        orow[lane + 32] = o1;
    }
}

extern "C" void kernel_launch(void* const* d_in, const int* in_sizes, int n_in,
                              void* d_out, int out_size, void* d_ws, size_t ws_size,
                              hipStream_t stream) {
    (void)in_sizes; (void)n_in; (void)out_size; (void)d_ws; (void)ws_size;
    const float* x     = (const float*)d_in[0];   // [8192, 256]
    const float* W     = (const float*)d_in[1];   // [4, 256]
    const float* b_lin = (const float*)d_in[2];   // [4]
    const float* bias  = (const float*)d_in[3];   // [4, 256]
    float* out         = (float*)d_out;           // [8192, 256]

    const int B = 8192;
    dcn_cross_wmma<<<B / ROWS_PER_BLOCK, 256, 0, stream>>>(x, W, b_lin, bias, out);
}